// S4Kernel_781684048339
// MI455X (gfx1250) — compile-verified
//
#include <hip/hip_runtime.h>
#include <hip/hip_bf16.h>

// S4 convolution kernel for MI455X (gfx1250).
//   Stage 0: per-(h,n) precompute of A*dt and v_ab = Bcat_a*Ccat_b*dt   (tiny)
//   Stage 1: k_f[h,l] pole sum + rank-1 feedback, stored as irfft-weighted
//            Re/Im planes (fp32, zero-padded K to a multiple of 4)
//   Stage 2: irfft as a real DFT GEMM on the f32 WMMA pipe:
//            out[h,t] = sum_l ( Xre[h,l]*cos(2*pi*l*t/L) + Xim[h,l]*(-sin(...)) )
//            using V_WMMA_F32_16X16X4_F32, with exact integer-mod-4096 phase
//            recurrences feeding v_cos_f32 / v_sin_f32 (inputs in revolutions).

#define S4_H    512
#define S4_N    64
#define S4_L    4096
#define S4_LF   2049          // L/2 + 1 rfft bins
#define S4_KP   2052          // LF padded to a multiple of 4 (WMMA K step)
#define S4_MASK 4095
#define S4_INVL (1.0f / 4096.0f)

typedef __attribute__((ext_vector_type(2))) float v2f;
typedef __attribute__((ext_vector_type(8))) float v8f;

// ---------------------------------------------------------------------------
// Stage 0: per-(h,n) constants -> pre[idx*10] =
//   { adt_re, adt_im, v00re, v00im, v01re, v01im, v10re, v10im, v11re, v11im }
// Bcat = [B, P], Ccat = [C, conj(P)], v_ab = Bcat_a * Ccat_b * dt
// ---------------------------------------------------------------------------
__global__ __launch_bounds__(256) void s4_precompute(
    const float* __restrict__ A_real, const float* __restrict__ A_imag,
    const float* __restrict__ B, const float* __restrict__ C,
    const float* __restrict__ P, const float* __restrict__ inv_dt,
    float* __restrict__ pre)
{
  int idx = blockIdx.x * 256 + threadIdx.x;          // 0 .. H*N-1
  if (idx >= S4_H * S4_N) return;
  int h = idx >> 6;

  float dt  = expf(inv_dt[h]);
  float Are = -expf(A_real[idx]);
  float Aim = A_imag[idx];

  float bre = B[2*idx+0], bim = B[2*idx+1];
  float cre = C[2*idx+0], cim = C[2*idx+1];
  float prr = P[2*idx+0], pri = P[2*idx+1];
  float qre = prr,        qim = -pri;                // conj(P)

  float* o = pre + (size_t)idx * 10;
  o[0] = Are * dt;  o[1] = Aim * dt;                 // A_dt
  o[2] = (bre*cre - bim*cim) * dt;  o[3] = (bre*cim + bim*cre) * dt;  // v00=B*C
  o[4] = (bre*qre - bim*qim) * dt;  o[5] = (bre*qim + bim*qre) * dt;  // v01=B*Q
  o[6] = (prr*cre - pri*cim) * dt;  o[7] = (prr*cim + pri*cre) * dt;  // v10=P*C
  o[8] = (prr*qre - pri*qim) * dt;  o[9] = (prr*qim + pri*qre) * dt;  // v11=P*Q
}

// ---------------------------------------------------------------------------
// Stage 1: spectrum. One thread per (h, l), l in [0, KP).
//   omega_l = e^{-2*pi*i*l/L}  (exact via hw trig on l/4096 revolutions)
//   z = 2(1-omega)/(1+omega)
//   r_ab = sum_n v_ab/(z - A_dt) + conj(v_ab)/(z - conj(A_dt))
//   k_f  = (r00 - r01*r10/(1+r11)) * 2/(1+omega)
// Stored with irfft weights folded in: w_l = 1/L at DC/Nyquist else 2/L.
// Nyquist (1+omega == 0 exactly): analytic limit k_f = Re(sum_n v00).
// ---------------------------------------------------------------------------
__global__ __launch_bounds__(256) void s4_kf(
    const float* __restrict__ pre,
    float* __restrict__ Xre, float* __restrict__ Xim)
{
  int l = blockIdx.x * 256 + threadIdx.x;
  int h = blockIdx.y;
  if (l >= S4_KP) return;
  size_t oidx = (size_t)h * S4_KP + l;

  if (l >= S4_LF) { Xre[oidx] = 0.0f; Xim[oidx] = 0.0f; return; }  // zero pad

  const float wl = (l == 0 || l == S4_L/2) ? (1.0f/S4_L) : (2.0f/S4_L);
  const float* pp = pre + (size_t)h * S4_N * 10;

  if (l == S4_L/2) {                                  // Nyquist analytic limit
    float s = 0.0f;
    for (int n = 0; n < S4_N; ++n) s += pp[n*10 + 2];
    Xre[oidx] = wl * s;
    Xim[oidx] = 0.0f;                                 // killed by sin(pi*t)=0
    return;
  }

  float u  = (float)l * S4_INVL;
  float cw = __builtin_amdgcn_cosf(u);                // cos(2*pi*l/L)
  float sw = __builtin_amdgcn_sinf(u);
  // omega = cw - i*sw ; den = 1+omega ; num = 1-omega ; z = 2*num/den
  float dre = 1.0f + cw, dim = -sw;
  float nre = 1.0f - cw, nim =  sw;
  float m2  = dre*dre + dim*dim;
  float im2 = __builtin_amdgcn_rcpf(m2);
  float zre = 2.0f * (nre*dre + nim*dim) * im2;
  float zim = 2.0f * (nim*dre - nre*dim) * im2;

  float r00re=0, r00im=0, r01re=0, r01im=0, r10re=0, r10im=0, r11re=0, r11im=0;
  for (int n = 0; n < S4_N; ++n) {
    float adre = pp[0], adim = pp[1];
    float wre  = zre - adre;
    float wi1  = zim - adim;                          // z - a
    float wi2  = zim + adim;                          // z - conj(a)
    float q1   = __builtin_amdgcn_rcpf(wre*wre + wi1*wi1);
    float q2   = __builtin_amdgcn_rcpf(wre*wre + wi2*wi2);
    float d_re =  wre * q1, d_im = -wi1 * q1;         // 1/(z-a)
    float e_re =  wre * q2, e_im = -wi2 * q2;         // 1/(z-conj(a))
    // r += v*d + conj(v)*e  (expanded, shared across the 4 ab combos)
    float sre = d_re + e_re;
    float sdi = e_im - d_im;
    float sim = d_im + e_im;
    float sdr = d_re - e_re;
    float vre, vim;
    vre = pp[2]; vim = pp[3];
    r00re += vre*sre + vim*sdi;  r00im += vre*sim + vim*sdr;
    vre = pp[4]; vim = pp[5];
    r01re += vre*sre + vim*sdi;  r01im += vre*sim + vim*sdr;
    vre = pp[6]; vim = pp[7];
    r10re += vre*sre + vim*sdi;  r10im += vre*sim + vim*sdr;
    vre = pp[8]; vim = pp[9];
    r11re += vre*sre + vim*sdi;  r11im += vre*sim + vim*sdr;
    pp += 10;
  }

  // k = r00 - r01*r10/(1+r11)
  float ere = 1.0f + r11re, eim = r11im;
  float iem = __builtin_amdgcn_rcpf(ere*ere + eim*eim);
  float tre = r01re*r10re - r01im*r10im;
  float tim = r01re*r10im + r01im*r10re;
  float fre = (tre*ere + tim*eim) * iem;
  float fim = (tim*ere - tre*eim) * iem;
  float kre = r00re - fre, kim = r00im - fim;
  // k_f = k * 2/(1+omega) = k * 2*conj(den)/|den|^2, then fold irfft weight
  float g = 2.0f * im2;
  Xre[oidx] = wl * (g * (kre*dre + kim*dim));
  Xim[oidx] = wl * (g * (kim*dre - kre*dim));
}

// ---------------------------------------------------------------------------
// Stage 2: irfft as DFT GEMM on V_WMMA_F32_16X16X4_F32.
// Each wave owns a 16-row (h) x 64-col (t) output tile = 4 accumulators.
// A fragments (16x4 f32): lane<16 holds K={4k,4k+1}, lane>=16 K={4k+2,4k+3}
// of row h0+(lane&15); B fragments (4x16): same per-lane K split, col lane&15.
// B values cos/-sin of 2*pi*(K*t mod 4096)/4096: integer phase recurrence
// p += (4t) mod 4096 per step (exact), K+1 via angle-addition with per-tile
// constants (ct, st) -> 2 trans ops per tile per step.
// ---------------------------------------------------------------------------
__global__ __launch_bounds__(256) void s4_dft_wmma(
    const float* __restrict__ Xre, const float* __restrict__ Xim,
    float* __restrict__ out)
{
  const int lane  = threadIdx.x & 31;
  const int wave  = threadIdx.x >> 5;
  const int gw    = blockIdx.x * 8 + wave;     // 0..2047
  const int mtile = gw & 31;                   // 32 row tiles of 16
  const int ngrp  = gw >> 5;                   // 64 col groups of 64
  const int h0    = mtile << 4;
  const int t0    = ngrp  << 6;
  const int klo   = (lane & 16) ? 2 : 0;
  const int nlane = lane & 15;

  const v2f* __restrict__ pa = (const v2f*)(Xre + (size_t)(h0 + nlane) * S4_KP + klo);
  const v2f* __restrict__ pb = (const v2f*)(Xim + (size_t)(h0 + nlane) * S4_KP + klo);

  int   t[4], p[4], dp[4];
  float ct[4], st[4];
#pragma unroll
  for (int j = 0; j < 4; ++j) {
    t[j]  = t0 + 16*j + nlane;
    p[j]  = (klo * t[j]) & S4_MASK;
    dp[j] = (4   * t[j]) & S4_MASK;
    float ut = (float)t[j] * S4_INVL;
    ct[j] = __builtin_amdgcn_cosf(ut);
    st[j] = __builtin_amdgcn_sinf(ut);
  }

  v8f acc[4] = {};

  for (int k = 0; k < S4_KP/4; ++k) {
    v2f are = pa[0];
    v2f aim = pb[0];
    pa += 2; pb += 2;                          // advance 4 floats
#pragma unroll
    for (int j = 0; j < 4; ++j) {
      float u0 = (float)p[j] * S4_INVL;
      float c0 = __builtin_amdgcn_cosf(u0);    // cos(2*pi*K*t/L), exact phase
      float s0 = __builtin_amdgcn_sinf(u0);
      v2f bc, bs;
      bc.x = c0;
      bc.y = c0*ct[j] - s0*st[j];              // cos at K+1 (angle addition)
      bs.x = -s0;
      bs.y = -(s0*ct[j] + c0*st[j]);           // -sin at K+1
      acc[j] = __builtin_amdgcn_wmma_f32_16x16x4_f32(
                   false, are, false, bc, (short)0, acc[j], false, false);
      acc[j] = __builtin_amdgcn_wmma_f32_16x16x4_f32(
                   false, aim, false, bs, (short)0, acc[j], false, false);
      p[j] = (p[j] + dp[j]) & S4_MASK;
    }
  }

  // C/D layout: VGPR v holds M=v (lanes 0-15) and M=v+8 (lanes 16-31), N=lane&15
  const int rbase = h0 + ((lane & 16) ? 8 : 0);
#pragma unroll
  for (int j = 0; j < 4; ++j) {
    float* o = out + (size_t)rbase * S4_L + (t0 + 16*j + nlane);
#pragma unroll
    for (int vg = 0; vg < 8; ++vg)
      o[(size_t)vg * S4_L] = acc[j][vg];
  }
}

// ---------------------------------------------------------------------------
// Workspace layout (floats):
//   [0,            H*KP)          Xre   (4.2 MB)
//   [H*KP,       2*H*KP)          Xim   (4.2 MB)
//   [2*H*KP, 2*H*KP + H*N*10)     pre   (1.3 MB)
// ---------------------------------------------------------------------------
extern "C" void kernel_launch(void* const* d_in, const int* in_sizes, int n_in,
                              void* d_out, int out_size, void* d_ws, size_t ws_size,
                              hipStream_t stream)
{
  (void)in_sizes; (void)n_in; (void)out_size; (void)ws_size;
  const float* A_real = (const float*)d_in[0];
  const float* A_imag = (const float*)d_in[1];
  const float* B      = (const float*)d_in[2];
  const float* C      = (const float*)d_in[3];
  const float* P      = (const float*)d_in[4];
  const float* inv_dt = (const float*)d_in[5];
  float* out = (float*)d_out;

  float* ws  = (float*)d_ws;
  float* Xre = ws;
  float* Xim = ws + (size_t)S4_H * S4_KP;
  float* pre = ws + (size_t)2 * S4_H * S4_KP;

  s4_precompute<<<dim3((S4_H*S4_N + 255)/256), dim3(256), 0, stream>>>(
      A_real, A_imag, B, C, P, inv_dt, pre);

  s4_kf<<<dim3((S4_KP + 255)/256, S4_H), dim3(256), 0, stream>>>(pre, Xre, Xim);

  // 2048 waves (32 mtiles x 64 ngroups), 8 waves per block
  s4_dft_wmma<<<dim3(256), dim3(256), 0, stream>>>(Xre, Xim, out);
}